// NTMMemory_60404420051214
// MI455X (gfx1250) — compile-verified
//
#include <hip/hip_runtime.h>
#include <math.h>
#include <stdint.h>

#define B_ 128
#define N_ 256
#define M_ 1024
#define EPS 1e-16f
#define NORM_EPS 1e-8f

typedef __attribute__((ext_vector_type(2))) float v2f;
typedef __attribute__((ext_vector_type(4))) float v4f;
typedef __attribute__((ext_vector_type(8))) float v8f;

#define TSTRIDE 68   // padded LDS row stride (floats); 272B = 17*16, 16B aligned

// ---------------------------------------------------------------------------
// Kernel 1: per (batch, 16-row tile) compute dot[b,n] = (mem+eps)·(k+eps)
// via V_WMMA_F32_16X16X4_F32, and sumsq[b,n] = ||mem+eps||^2 via VALU.
// Staging uses CDNA5 GLOBAL_LOAD_ASYNC_TO_LDS_B128 with double-buffered LDS
// tiles (ASYNCcnt pipelining). One wave per block; EXEC all-ones for WMMA.
// ---------------------------------------------------------------------------
__global__ __launch_bounds__(32) void k_dot_norm(
    const float* __restrict__ memory, const float* __restrict__ kvec,
    float* __restrict__ dotOut, float* __restrict__ ssOut)
{
    __shared__ float kLDS[M_];                // k row (+eps)
    __shared__ float tile[2][16 * TSTRIDE];   // double-buffered 16x64 raw tiles

    const int b    = blockIdx.y;
    const int n0   = blockIdx.x * 16;
    const int lane = threadIdx.x;          // 0..31
    const int r    = lane & 15;            // A-matrix row
    const int koff = (lane >> 4) * 2;      // A-matrix K offset: 0 or 2

    // Stage k[b] + eps into LDS (coalesced float4, one-time)
    const float* kb = kvec + (size_t)b * M_;
    for (int i = lane; i < M_ / 4; i += 32) {
        v4f kv = *(const v4f*)(kb + 4 * i);
        kLDS[4*i + 0] = kv.x + EPS;
        kLDS[4*i + 1] = kv.y + EPS;
        kLDS[4*i + 2] = kv.z + EPS;
        kLDS[4*i + 3] = kv.w + EPS;
    }
    __syncthreads();

    const float* mb = memory + ((size_t)b * N_ + n0) * M_;
    const int srow = lane >> 4;            // staging row within pair: 0/1
    const int scol = (lane & 15) * 4;      // staging column (floats)

    // Issue 8 async b128 copies for one 16x64 tile (512B per instruction)
    auto issue_tile = [&](int buf, int m0) {
        #pragma unroll
        for (int p = 0; p < 8; ++p) {
            int rr = 2 * p + srow;
            uint32_t lds = (uint32_t)(uintptr_t)&tile[buf][rr * TSTRIDE + scol];
            uint64_t ga  = (uint64_t)(uintptr_t)(mb + (size_t)rr * M_ + m0 + scol);
            asm volatile("global_load_async_to_lds_b128 %0, %1, off"
                         : : "v"(lds), "v"(ga) : "memory");
        }
    };

    v8f acc = {};
    float ss = 0.0f;

    issue_tile(0, 0);
    for (int c = 0; c < M_ / 64; ++c) {
        const int m0 = c * 64;
        if (c + 1 < M_ / 64) {
            issue_tile((c + 1) & 1, m0 + 64);
            asm volatile("s_wait_asynccnt 0x8" ::: "memory"); // current tile landed
        } else {
            asm volatile("s_wait_asynccnt 0x0" ::: "memory");
        }

        const float* tb = &tile[c & 1][0];
        #pragma unroll
        for (int kk = 0; kk < 64; kk += 4) {
            v2f a = *(const v2f*)&tb[r * TSTRIDE + kk + koff];
            v2f av; av.x = a.x + EPS; av.y = a.y + EPS;   // mem + eps
            v2f bb;
            bb.x = kLDS[m0 + kk + koff];
            bb.y = kLDS[m0 + kk + koff + 1];
            ss += av.x * av.x + av.y * av.y;
            acc = __builtin_amdgcn_wmma_f32_16x16x4_f32(
                false, av, false, bb, (short)0, acc, false, false);
        }
    }

    // Extract dot: VGPR v of lanes 0-15 holds row v, lanes 16-31 row 8+v.
    if (lane == 0) {
        #pragma unroll
        for (int v = 0; v < 8; ++v)
            dotOut[(size_t)b * N_ + n0 + v] = acc[v];
    } else if (lane == 16) {
        #pragma unroll
        for (int v = 0; v < 8; ++v)
            dotOut[(size_t)b * N_ + n0 + 8 + v] = acc[v];
    }

    // sumsq: lane L covered half of row r; combine with the other half-wave.
    ss += __shfl_xor(ss, 16, 32);
    if (lane < 16)
        ssOut[(size_t)b * N_ + n0 + lane] = ss;
}

// ---------------------------------------------------------------------------
// Kernel 2: per batch: cosine -> softmax -> interpolate -> shift -> sharpen
// ---------------------------------------------------------------------------
__global__ __launch_bounds__(256) void k_weights(
    const float* __restrict__ kvec, const float* __restrict__ beta,
    const float* __restrict__ g, const float* __restrict__ s,
    const float* __restrict__ gamma, const float* __restrict__ w_prev,
    const float* __restrict__ dotIn, const float* __restrict__ ssIn,
    float* __restrict__ wOut)
{
    __shared__ float red[256];
    __shared__ float wgS[256];
    const int b = blockIdx.x;
    const int t = threadIdx.x;

    // ||k + eps||
    float a0 = 0.0f;
    for (int m = t; m < M_; m += 256) {
        float kv = kvec[(size_t)b * M_ + m] + EPS;
        a0 += kv * kv;
    }
    red[t] = a0; __syncthreads();
    for (int off = 128; off > 0; off >>= 1) {
        if (t < off) red[t] += red[t + off];
        __syncthreads();
    }
    const float knorm = fmaxf(sqrtf(red[0]), NORM_EPS);
    __syncthreads();

    const float mnorm = fmaxf(sqrtf(ssIn[(size_t)b * N_ + t]), NORM_EPS);
    const float x = beta[b] * (dotIn[(size_t)b * N_ + t] / (mnorm * knorm));

    // softmax over N (max-subtracted)
    red[t] = x; __syncthreads();
    for (int off = 128; off > 0; off >>= 1) {
        if (t < off) red[t] = fmaxf(red[t], red[t + off]);
        __syncthreads();
    }
    const float mx = red[0]; __syncthreads();
    const float ex = expf(x - mx);
    red[t] = ex; __syncthreads();
    for (int off = 128; off > 0; off >>= 1) {
        if (t < off) red[t] += red[t + off];
        __syncthreads();
    }
    const float wc = ex / red[0]; __syncthreads();

    // interpolate
    const float gg = g[b];
    const float wg = gg * wc + (1.0f - gg) * w_prev[(size_t)b * N_ + t];

    // circular 3-tap shift
    wgS[t] = wg; __syncthreads();
    const float wt = s[b * 3 + 0] * wgS[(t + N_ - 1) & (N_ - 1)]
                   + s[b * 3 + 1] * wg
                   + s[b * 3 + 2] * wgS[(t + 1) & (N_ - 1)];

    // sharpen
    const float wp = powf(wt, gamma[b]);
    red[t] = wp; __syncthreads();
    for (int off = 128; off > 0; off >>= 1) {
        if (t < off) red[t] += red[t + off];
        __syncthreads();
    }
    wOut[(size_t)b * N_ + t] = wp / (red[0] + EPS);
}

// ---------------------------------------------------------------------------
// Kernel 3: fused write + read. new_memory stores are non-temporal so the
// `memory` lines warmed by K1 stay resident in the 192MB L2 for this pass.
// ---------------------------------------------------------------------------
__global__ __launch_bounds__(256) void k_write_read(
    const float* __restrict__ memory, const float* __restrict__ e,
    const float* __restrict__ a, const float* __restrict__ w,
    float* __restrict__ newMem, float* __restrict__ readOut)
{
    __shared__ float wS[N_];
    const int b = blockIdx.y;
    const int m = blockIdx.x * 256 + threadIdx.x;

    for (int n = threadIdx.x; n < N_; n += 256)
        wS[n] = w[(size_t)b * N_ + n];
    __syncthreads();

    const float em = e[(size_t)b * M_ + m];
    const float am = a[(size_t)b * M_ + m];

    const float* mp = memory + (size_t)b * N_ * M_ + m;
    float*       np = newMem + (size_t)b * N_ * M_ + m;

    float acc = 0.0f;
    #pragma unroll 4
    for (int n = 0; n < N_; ++n) {
        const float wn = wS[n];
        const float v  = mp[(size_t)n * M_];
        const float nm = v * (1.0f - wn * em) + wn * am;
        __builtin_nontemporal_store(nm, &np[(size_t)n * M_]);
        acc += wn * nm;
    }
    readOut[(size_t)b * M_ + m] = acc;
}

// ---------------------------------------------------------------------------
extern "C" void kernel_launch(void* const* d_in, const int* in_sizes, int n_in,
                              void* d_out, int out_size, void* d_ws, size_t ws_size,
                              hipStream_t stream)
{
    const float* memory = (const float*)d_in[0];
    const float* k      = (const float*)d_in[1];
    const float* beta   = (const float*)d_in[2];
    const float* g      = (const float*)d_in[3];
    const float* s      = (const float*)d_in[4];
    const float* gamma  = (const float*)d_in[5];
    const float* w_prev = (const float*)d_in[6];
    const float* e      = (const float*)d_in[7];
    const float* a      = (const float*)d_in[8];

    // outputs concatenated: read (B,M), new_memory (B,N,M), w (B,N)
    float* out_read   = (float*)d_out;
    float* out_newmem = out_read + (size_t)B_ * M_;
    float* out_w      = out_newmem + (size_t)B_ * N_ * M_;

    float* dotWs = (float*)d_ws;            // B*N floats
    float* ssWs  = dotWs + (size_t)B_ * N_; // B*N floats

    k_dot_norm<<<dim3(N_ / 16, B_), 32, 0, stream>>>(memory, k, dotWs, ssWs);
    k_weights<<<dim3(B_), 256, 0, stream>>>(k, beta, g, s, gamma, w_prev,
                                            dotWs, ssWs, out_w);
    k_write_read<<<dim3(M_ / 256, B_), 256, 0, stream>>>(memory, e, a, out_w,
                                                         out_newmem, out_read);
}